// MPNN_layer_50027779064047
// MI455X (gfx1250) — compile-verified
//
#include <hip/hip_runtime.h>
#include <hip/hip_bf16.h>

#define DN 128
#define DE 64
#define DO 128
#define WIN (2*DN + DE)   // 320

typedef float v2f __attribute__((ext_vector_type(2)));
typedef float v8f __attribute__((ext_vector_type(8)));

// ---------------------------------------------------------------------------
// Kernel 1: P[n, 0:256] = X[n,:] @ [W_src^T | W_dst^T]
//   P[n, o]       = sum_k X[n,k] * W[o, k]        (o in 0..127  -> src proj)
//   P[n, 128+o]   = sum_k X[n,k] * W[o, 128+k]    (o in 0..127  -> dst proj)
// One wave computes a 16(nodes) x 16(outs) tile via V_WMMA_F32_16X16X4_F32.
// ---------------------------------------------------------------------------
__global__ void node_proj_kernel(const float* __restrict__ X,   // [N, DN]
                                 const float* __restrict__ W,   // [DO, WIN]
                                 float* __restrict__ P,         // [N, 2*DO]
                                 int N)
{
    const int wave = (blockIdx.x * blockDim.x + threadIdx.x) >> 5;
    const int lane = threadIdx.x & 31;
    const int tilesO = (2 * DO) / 16;          // 16 output tiles
    const int tn = wave / tilesO;
    const int to = wave % tilesO;
    const int n0 = tn * 16;
    if (n0 >= N) return;                       // wave-uniform: EXEC stays full
    const int o0 = to * 16;

    const int l15 = lane & 15;
    const int hi  = lane >> 4;                 // 0: K={0,1}, 1: K={2,3}

    // A fragment row (node), clamped for the tail tile (stores are guarded)
    int arow = n0 + l15; if (arow >= N) arow = N - 1;
    // B fragment row: output column o0+l15 maps to a row of W (transposed read)
    const int ocol = o0 + l15;
    const int wrow = (ocol < DO) ? ocol : (ocol - DO);
    const int koff = (ocol < DO) ? 0 : DN;     // W_src vs W_dst block

    const float* Arow = X + (size_t)arow * DN;
    const float* Brow = W + (size_t)wrow * WIN + koff;

    v8f acc = {};
    #pragma unroll
    for (int k0 = 0; k0 < DN; k0 += 4) {
        v2f a = *(const v2f*)(Arow + k0 + 2 * hi);   // A[M=l15][K=k0+2hi{,+1}]
        v2f b = *(const v2f*)(Brow + k0 + 2 * hi);   // B[K=k0+2hi{,+1}][N=l15]
        acc = __builtin_amdgcn_wmma_f32_16x16x4_f32(
            /*neg_a=*/false, a, /*neg_b=*/false, b,
            /*c_mod=*/(short)0, acc, /*reuse_a=*/false, /*reuse_b=*/false);
    }

    // D layout: VGPR r -> M = r + 8*hi, N = l15
    float* Pt = P + o0 + l15;
    #pragma unroll
    for (int r = 0; r < 8; ++r) {
        const int m = n0 + r + 8 * hi;
        if (m < N) Pt[(size_t)m * (2 * DO)] = acc[r];
    }
}

// ---------------------------------------------------------------------------
// Kernel 2: per-edge message part + scatter-add.
//   Pe[e, o] = sum_k EF[e,k] * W[o, 256+k]   (K = 64 -> 16 WMMAs)
//   sums[dst[e], o] += Pe[e, o] + P_src[src[e], o]
//   deg[dst[e]]     += 1        (once per edge, done by the to==0 wave)
// One wave computes a 16(edges) x 16(outs) tile.
// ---------------------------------------------------------------------------
__global__ void edge_msg_kernel(const float* __restrict__ EF,   // [E, DE]
                                const float* __restrict__ W,    // [DO, WIN]
                                const float* __restrict__ P,    // [N, 2*DO]
                                const int*   __restrict__ src,  // [E]
                                const int*   __restrict__ dst,  // [E]
                                float* __restrict__ sums,       // [N, DO]
                                float* __restrict__ deg,        // [N]
                                int E)
{
    const int wave = (blockIdx.x * blockDim.x + threadIdx.x) >> 5;
    const int lane = threadIdx.x & 31;
    const int tilesO = DO / 16;                // 8 output tiles
    const int te = wave / tilesO;
    const int to = wave % tilesO;
    const int e0 = te * 16;
    if (e0 >= E) return;                       // wave-uniform
    const int o0 = to * 16;

    const int l15 = lane & 15;
    const int hi  = lane >> 4;

    int erow = e0 + l15; if (erow >= E) erow = E - 1;
    const float* Arow = EF + (size_t)erow * DE;
    const float* Brow = W + (size_t)(o0 + l15) * WIN + 2 * DN;  // W_e block

    v8f acc = {};
    #pragma unroll
    for (int k0 = 0; k0 < DE; k0 += 4) {
        v2f a = *(const v2f*)(Arow + k0 + 2 * hi);
        v2f b = *(const v2f*)(Brow + k0 + 2 * hi);
        acc = __builtin_amdgcn_wmma_f32_16x16x4_f32(
            false, a, false, b, (short)0, acc, false, false);
    }

    // Element (r, lane): edge e0 + r + 8*hi, output o0 + l15
    #pragma unroll
    for (int r = 0; r < 8; ++r) {
        const int e = e0 + r + 8 * hi;
        if (e < E) {
            const int s = src[e];              // broadcast across lanes 0-15 / 16-31
            const int d = dst[e];
            const float val = acc[r] + P[(size_t)s * (2 * DO) + o0 + l15];
            atomicAdd(&sums[(size_t)d * DO + o0 + l15], val);
        }
    }

    // Count each edge exactly once: only the to==0 wave, lanes 0..15.
    if (to == 0 && hi == 0) {
        const int e = e0 + l15;
        if (e < E) atomicAdd(&deg[dst[e]], 1.0f);
    }
}

// ---------------------------------------------------------------------------
// Kernel 3: h[n,o] = relu(deg>0 ? sums[n,o]/deg + P_dst[n,o] + b[o] : X[n,o])
// ---------------------------------------------------------------------------
__global__ void finalize_kernel(const float* __restrict__ X,
                                const float* __restrict__ P,    // [N, 2*DO]
                                const float* __restrict__ bias, // [DO]
                                const float* __restrict__ sums, // [N, DO]
                                const float* __restrict__ deg,  // [N]
                                float* __restrict__ out,        // [N, DO]
                                int N)
{
    const int i = blockIdx.x * blockDim.x + threadIdx.x;
    if (i >= N * DO) return;
    const int n = i / DO;
    const int o = i - n * DO;
    const float dg = deg[n];
    float h;
    if (dg > 0.0f) {
        h = sums[i] / dg + P[(size_t)n * (2 * DO) + DO + o] + bias[o];
    } else {
        h = X[i];
    }
    out[i] = fmaxf(h, 0.0f);
}

// ---------------------------------------------------------------------------
extern "C" void kernel_launch(void* const* d_in, const int* in_sizes, int n_in,
                              void* d_out, int out_size, void* d_ws, size_t ws_size,
                              hipStream_t stream)
{
    const float* node_feats = (const float*)d_in[0];
    const float* edge_feats = (const float*)d_in[1];
    const float* W          = (const float*)d_in[2];
    const float* bias       = (const float*)d_in[3];
    const int*   src        = (const int*)d_in[4];
    const int*   dst        = (const int*)d_in[5];

    const int N = in_sizes[0] / DN;
    const int E = in_sizes[1] / DE;

    // Workspace layout: P[N*256] | sums[N*128] | deg[N]
    float* P    = (float*)d_ws;
    float* sums = P + (size_t)N * (2 * DO);
    float* deg  = sums + (size_t)N * DO;

    // Atomic accumulators must start at zero every launch (graph-replay safe).
    hipMemsetAsync(sums, 0, ((size_t)N * DO + (size_t)N) * sizeof(float), stream);

    {   // node projection: ceil(N/16) * 16 tiles, 1 wave per tile
        const int tilesN = (N + 15) / 16;
        const long long waves = (long long)tilesN * ((2 * DO) / 16);
        const int block = 256;
        const long long grid = (waves * 32 + block - 1) / block;
        node_proj_kernel<<<(int)grid, block, 0, stream>>>(node_feats, W, P, N);
    }
    {   // edge messages + scatter: ceil(E/16) * 8 tiles, 1 wave per tile
        const int tilesE = (E + 15) / 16;
        const long long waves = (long long)tilesE * (DO / 16);
        const int block = 256;
        const long long grid = (waves * 32 + block - 1) / block;
        edge_msg_kernel<<<(int)grid, block, 0, stream>>>(edge_feats, W, P, src, dst,
                                                         sums, deg, E);
    }
    {   // finalize
        const int total = N * DO;
        const int block = 256;
        const int grid = (total + block - 1) / block;
        finalize_kernel<<<grid, block, 0, stream>>>(node_feats, P, bias, sums, deg,
                                                    (float*)d_out, N);
    }
}